// MIGGT_NodeClassifier_2568390443300
// MI455X (gfx1250) — compile-verified
//
#include <hip/hip_runtime.h>
#include <hip/hip_bf16.h>

#define N_NODES 30000
#define N_EDGES 960000
#define HID 256
#define KT 768
#define KV 512
#define SLEN 16   // 1 + S(15)
#define ALPHA_C 0.1f
#define BETA_C 0.9f

typedef float v2f __attribute__((ext_vector_type(2)));
typedef float v8f __attribute__((ext_vector_type(8)));

#define WMMA_F32(a, b, c) \
  __builtin_amdgcn_wmma_f32_16x16x4_f32(false, (a), false, (b), (short)0, (c), false, false)

static __device__ inline v8f vzero8() { return (v8f)0.0f; }

// ---------------------------------------------------------------------------
// Encoder: X = leaky(bn(text@tW+tb)) + leaky(bn(visual@vW+vb))
// One block = 16 node rows x 256 cols. 8 waves; wave w covers cols [32w,32w+32).
// ---------------------------------------------------------------------------
template <int KDIM>
__device__ inline void gemm_block(const float* __restrict__ feat,
                                  const float* __restrict__ W,
                                  float* __restrict__ As, int m0, v8f acc[2]) {
  const int lane = threadIdx.x & 31;
  const int half = lane >> 4;
  const int l15  = lane & 15;
  const int koff = half * 2;
  const int n0   = (threadIdx.x >> 5) * 32;
  const int lr   = threadIdx.x >> 4;   // A-loader row 0..15
  const int lc4  = threadIdx.x & 15;   // A-loader float4 col 0..15
  for (int k0 = 0; k0 < KDIM; k0 += 64) {
    __syncthreads();
    float4 v = *(const float4*)(feat + (size_t)(m0 + lr) * KDIM + k0 + lc4 * 4);
    float* dp = As + lr * 68 + lc4 * 4;
    dp[0] = v.x; dp[1] = v.y; dp[2] = v.z; dp[3] = v.w;
    __syncthreads();
#pragma unroll
    for (int kk = 0; kk < 64; kk += 4) {
      v2f a;
      a.x = As[l15 * 68 + kk + koff];
      a.y = As[l15 * 68 + kk + koff + 1];
      const int kb = k0 + kk + koff;
#pragma unroll
      for (int t = 0; t < 2; ++t) {
        const int n = n0 + t * 16 + l15;
        v2f b;
        b.x = W[(size_t)kb * HID + n];
        b.y = W[(size_t)(kb + 1) * HID + n];
        acc[t] = WMMA_F32(a, b, acc[t]);
      }
    }
  }
}

__global__ __launch_bounds__(256) void encoder_kernel(
    const float* __restrict__ text, const float* __restrict__ visual,
    const float* __restrict__ tW, const float* __restrict__ tb,
    const float* __restrict__ tg, const float* __restrict__ tbt,
    const float* __restrict__ ta,
    const float* __restrict__ vW, const float* __restrict__ vb,
    const float* __restrict__ vg, const float* __restrict__ vbt,
    const float* __restrict__ va, float* __restrict__ X) {
  __shared__ float As[16 * 68];
  const int m0   = blockIdx.x * 16;
  const int lane = threadIdx.x & 31;
  const int half = lane >> 4;
  const int l15  = lane & 15;
  const int n0   = (threadIdx.x >> 5) * 32;
  const float inv_bn = rsqrtf(1.0f + 1e-5f);

  v8f accT[2] = {vzero8(), vzero8()};
  v8f accV[2] = {vzero8(), vzero8()};
  gemm_block<KT>(text, tW, As, m0, accT);
  gemm_block<KV>(visual, vW, As, m0, accV);

  const float aT = ta[0], aV = va[0];
#pragma unroll
  for (int t = 0; t < 2; ++t) {
    const int n = n0 + t * 16 + l15;
    const float gT = tg[n] * inv_bn, bT = tb[n], btT = tbt[n];
    const float gV = vg[n] * inv_bn, bV = vb[n], btV = vbt[n];
#pragma unroll
    for (int r = 0; r < 8; ++r) {
      const int m = r + half * 8;
      float ht = (accT[t][r] + bT) * gT + btT;
      ht = (ht >= 0.0f) ? ht : aT * ht;
      float hv = (accV[t][r] + bV) * gV + btV;
      hv = (hv >= 0.0f) ? hv : aV * hv;
      X[(size_t)(m0 + m) * HID + n] = ht + hv;
    }
  }
}

// ---------------------------------------------------------------------------
// Graph diffusion helpers
// ---------------------------------------------------------------------------
__global__ __launch_bounds__(256) void zero_kernel(float* __restrict__ p, int n) {
  int i = blockIdx.x * 256 + threadIdx.x;
  if (i < n) p[i] = 0.0f;
}

__global__ __launch_bounds__(256) void deg_kernel(const int* __restrict__ dst,
                                                  float* __restrict__ deg) {
  int e = blockIdx.x * 256 + threadIdx.x;
  if (e < N_EDGES)
    __hip_atomic_fetch_add(&deg[dst[e]], 1.0f, __ATOMIC_RELAXED,
                           __HIP_MEMORY_SCOPE_AGENT);
}

__global__ __launch_bounds__(256) void norm_kernel(const int* __restrict__ src,
                                                   const int* __restrict__ dst,
                                                   const float* __restrict__ deg,
                                                   float* __restrict__ nrm) {
  int e = blockIdx.x * 256 + threadIdx.x;
  if (e < N_EDGES) {
    float ds = fmaxf(deg[src[e]], 1.0f);
    float dd = fmaxf(deg[dst[e]], 1.0f);
    nrm[e] = rsqrtf(ds * dd);
  }
}

// One thread per (edge, float4 chunk): 64 chunks per edge. h fits in L2.
__global__ __launch_bounds__(256) void scatter_kernel(
    const float* __restrict__ h, const int* __restrict__ src,
    const int* __restrict__ dst, const float* __restrict__ nrm,
    float* __restrict__ tmp) {
  int tid = blockIdx.x * 256 + threadIdx.x;
  int e = tid >> 6;
  int c = tid & 63;
  if (e >= N_EDGES) return;
  float nv = nrm[e];
  int s = src[e], d = dst[e];
  float4 v = *(const float4*)(h + (size_t)s * HID + c * 4);
  float* base = tmp + (size_t)d * HID + c * 4;
  __hip_atomic_fetch_add(base + 0, v.x * nv, __ATOMIC_RELAXED, __HIP_MEMORY_SCOPE_AGENT);
  __hip_atomic_fetch_add(base + 1, v.y * nv, __ATOMIC_RELAXED, __HIP_MEMORY_SCOPE_AGENT);
  __hip_atomic_fetch_add(base + 2, v.z * nv, __ATOMIC_RELAXED, __HIP_MEMORY_SCOPE_AGENT);
  __hip_atomic_fetch_add(base + 3, v.w * nv, __ATOMIC_RELAXED, __HIP_MEMORY_SCOPE_AGENT);
}

__global__ __launch_bounds__(256) void axpy_kernel(const float* __restrict__ X,
                                                   const float* __restrict__ tmp,
                                                   float* __restrict__ h,
                                                   float scale) {
  int i = blockIdx.x * 256 + threadIdx.x;
  h[i] = (ALPHA_C * X[i] + BETA_C * tmp[i]) * scale;
}

// ---------------------------------------------------------------------------
// MHA: one block per node. LDS tiles: mem, Q(=out), K, V (16x256 stride 257)
// plus 4x16x16 scores. All GEMMs use v_wmma_f32_16x16x4_f32.
// ---------------------------------------------------------------------------
#define TSTRIDE 257

__device__ inline void qkv_gemm(const float* __restrict__ A_lds,
                                const float* __restrict__ W,
                                const float* __restrict__ bias,
                                float* __restrict__ dstS) {
  const int lane = threadIdx.x & 31;
  const int half = lane >> 4;
  const int l15  = lane & 15;
  const int koff = half * 2;
  const int n0   = (threadIdx.x >> 5) * 32;
  v8f acc[2] = {vzero8(), vzero8()};
  for (int k = 0; k < HID; k += 4) {
    v2f a;
    a.x = A_lds[l15 * TSTRIDE + k + koff];
    a.y = A_lds[l15 * TSTRIDE + k + koff + 1];
    const int kb = k + koff;
#pragma unroll
    for (int t = 0; t < 2; ++t) {
      const int n = n0 + t * 16 + l15;
      v2f b;
      b.x = W[(size_t)kb * HID + n];
      b.y = W[(size_t)(kb + 1) * HID + n];
      acc[t] = WMMA_F32(a, b, acc[t]);
    }
  }
#pragma unroll
  for (int t = 0; t < 2; ++t) {
    const int n = n0 + t * 16 + l15;
    const float bv = bias[n];
#pragma unroll
    for (int r = 0; r < 8; ++r) {
      const int m = r + half * 8;
      dstS[m * TSTRIDE + n] = acc[t][r] + bv;
    }
  }
}

__global__ __launch_bounds__(256) void mha_kernel(
    const float* __restrict__ comb, const int* __restrict__ memidx,
    const float* __restrict__ Wq, const float* __restrict__ bq,
    const float* __restrict__ Wk, const float* __restrict__ bk,
    const float* __restrict__ Wv, const float* __restrict__ bv,
    const float* __restrict__ Wo, const float* __restrict__ bo,
    const float* __restrict__ Wc, const float* __restrict__ bc,
    float* __restrict__ zOut, float* __restrict__ fOut,
    float* __restrict__ logits) {
  extern __shared__ float smem[];
  float* memS = smem;                     // 16*257
  float* qS   = memS + 16 * TSTRIDE;      // reused as outS after scores
  float* kS   = qS + 16 * TSTRIDE;
  float* vS   = kS + 16 * TSTRIDE;
  float* sS   = vS + 16 * TSTRIDE;        // 4*16*16 scores; later row0 cache

  const int node = blockIdx.x;
  const int lane = threadIdx.x & 31;
  const int wave = threadIdx.x >> 5;
  const int half = lane >> 4;
  const int l15  = lane & 15;
  const int koff = half * 2;
  const int n0   = wave * 32;

  // Load mem tile: row 0 = node itself, rows 1..15 = memory neighbors.
#pragma unroll
  for (int i = 0; i < 4; ++i) {
    int idx = threadIdx.x + i * 256;  // float4 index, 0..1023
    int r   = idx >> 6;
    int c4  = idx & 63;
    int id  = (r == 0) ? node : memidx[node * 15 + r - 1];
    float4 v = *(const float4*)(comb + (size_t)id * HID + c4 * 4);
    float* dp = memS + r * TSTRIDE + c4 * 4;
    dp[0] = v.x; dp[1] = v.y; dp[2] = v.z; dp[3] = v.w;
  }
  __syncthreads();

  // Q, K, V projections (each 16x256x256).
  qkv_gemm(memS, Wq, bq, qS);
  qkv_gemm(memS, Wk, bk, kS);
  qkv_gemm(memS, Wv, bv, vS);
  __syncthreads();

  // Scores: waves 0..3, one head each. scores = Q_h (16x64) @ K_h^T / 8.
  if (wave < 4) {
    const int hh = wave;
    v8f acc = vzero8();
    for (int kk = 0; kk < 64; kk += 4) {
      v2f a;
      a.x = qS[l15 * TSTRIDE + hh * 64 + kk + koff];
      a.y = qS[l15 * TSTRIDE + hh * 64 + kk + koff + 1];
      v2f b;
      b.x = kS[l15 * TSTRIDE + hh * 64 + kk + koff];
      b.y = kS[l15 * TSTRIDE + hh * 64 + kk + koff + 1];
      acc = WMMA_F32(a, b, acc);
    }
#pragma unroll
    for (int r = 0; r < 8; ++r) {
      const int m = r + half * 8;
      sS[hh * 256 + m * 16 + l15] = acc[r] * 0.125f;
    }
  }
  __syncthreads();

  // Softmax over last dim: 64 rows of 16.
  if (threadIdx.x < 64) {
    float* rowp = sS + threadIdx.x * 16;
    float mx = rowp[0];
#pragma unroll
    for (int j = 1; j < 16; ++j) mx = fmaxf(mx, rowp[j]);
    float e[16];
    float sum = 0.0f;
#pragma unroll
    for (int j = 0; j < 16; ++j) { e[j] = __expf(rowp[j] - mx); sum += e[j]; }
    float inv = 1.0f / sum;
#pragma unroll
    for (int j = 0; j < 16; ++j) rowp[j] = e[j] * inv;
  }
  __syncthreads();

  // out = attn @ V (per head). Wave w: head w>>1, two 16-col tiles of that head.
  {
    const int hh = wave >> 1;
    const int ntb = (wave & 1) * 2;
    v8f acc[2] = {vzero8(), vzero8()};
#pragma unroll
    for (int kk = 0; kk < 16; kk += 4) {
      v2f a;
      a.x = sS[hh * 256 + l15 * 16 + kk + koff];
      a.y = sS[hh * 256 + l15 * 16 + kk + koff + 1];
      const int kb = kk + koff;
#pragma unroll
      for (int t = 0; t < 2; ++t) {
        const int n = hh * 64 + (ntb + t) * 16 + l15;
        v2f b;
        b.x = vS[kb * TSTRIDE + n];
        b.y = vS[(kb + 1) * TSTRIDE + n];
        acc[t] = WMMA_F32(a, b, acc[t]);
      }
    }
    __syncthreads();  // all reads of qS (scores step) done; safe to overwrite
#pragma unroll
    for (int t = 0; t < 2; ++t) {
      const int gn = hh * 64 + (ntb + t) * 16 + l15;
#pragma unroll
      for (int r = 0; r < 8; ++r) {
        const int m = r + half * 8;
        qS[m * TSTRIDE + gn] = acc[t][r];  // qS now holds attn output
      }
    }
  }
  __syncthreads();

  // z = out @ Wo + bo + mem  -> global; row 0 also -> final_h + sS cache.
  {
    v8f acc[2] = {vzero8(), vzero8()};
    for (int k = 0; k < HID; k += 4) {
      v2f a;
      a.x = qS[l15 * TSTRIDE + k + koff];
      a.y = qS[l15 * TSTRIDE + k + koff + 1];
      const int kb = k + koff;
#pragma unroll
      for (int t = 0; t < 2; ++t) {
        const int n = n0 + t * 16 + l15;
        v2f b;
        b.x = Wo[(size_t)kb * HID + n];
        b.y = Wo[(size_t)(kb + 1) * HID + n];
        acc[t] = WMMA_F32(a, b, acc[t]);
      }
    }
#pragma unroll
    for (int t = 0; t < 2; ++t) {
      const int n = n0 + t * 16 + l15;
      const float bvv = bo[n];
#pragma unroll
      for (int r = 0; r < 8; ++r) {
        const int m = r + half * 8;
        float z = acc[t][r] + bvv + memS[m * TSTRIDE + n];
        zOut[((size_t)node * SLEN + m) * HID + n] = z;
        if (m == 0) {
          fOut[(size_t)node * HID + n] = z;
          sS[n] = z;
        }
      }
    }
  }
  __syncthreads();

  // Classifier: logits = final_h @ Wc + bc (256 -> 40).
  if (threadIdx.x < 40) {
    float acc = bc[threadIdx.x];
    for (int k = 0; k < HID; ++k) acc += sS[k] * Wc[k * 40 + threadIdx.x];
    logits[(size_t)node * 40 + threadIdx.x] = acc;
  }
}

// ---------------------------------------------------------------------------
extern "C" void kernel_launch(void* const* d_in, const int* in_sizes, int n_in,
                              void* d_out, int out_size, void* d_ws,
                              size_t ws_size, hipStream_t stream) {
  const float* text   = (const float*)d_in[0];
  const float* visual = (const float*)d_in[1];
  const int*   edge   = (const int*)d_in[2];
  const int*   memidx = (const int*)d_in[3];
  const float* tW  = (const float*)d_in[4];
  const float* tb  = (const float*)d_in[5];
  const float* tg  = (const float*)d_in[6];
  const float* tbt = (const float*)d_in[7];
  const float* ta  = (const float*)d_in[8];
  const float* vW  = (const float*)d_in[9];
  const float* vb  = (const float*)d_in[10];
  const float* vg  = (const float*)d_in[11];
  const float* vbt = (const float*)d_in[12];
  const float* va  = (const float*)d_in[13];
  const float* Wq  = (const float*)d_in[14];
  const float* bq  = (const float*)d_in[15];
  const float* Wk  = (const float*)d_in[16];
  const float* bk  = (const float*)d_in[17];
  const float* Wv  = (const float*)d_in[18];
  const float* bvp = (const float*)d_in[19];
  const float* Wo  = (const float*)d_in[20];
  const float* bo  = (const float*)d_in[21];
  const float* Wc  = (const float*)d_in[22];
  const float* bc  = (const float*)d_in[23];

  const int* src = edge;
  const int* dst = edge + N_EDGES;

  float* ws  = (float*)d_ws;
  float* X   = ws;                              // 7.68M
  float* h   = X + (size_t)N_NODES * HID;       // 7.68M
  float* tmp = h + (size_t)N_NODES * HID;       // 7.68M
  float* deg = tmp + (size_t)N_NODES * HID;     // 30000
  float* nrm = deg + N_NODES;                   // 960000  (~96 MB total)

  float* out_logits = (float*)d_out;
  float* out_final  = out_logits + (size_t)N_NODES * 40;
  float* out_z      = out_final + (size_t)N_NODES * HID;

  // 1) Fused encoders (mgdcf is linear => diffuse the sum once).
  encoder_kernel<<<N_NODES / 16, 256, 0, stream>>>(
      text, visual, tW, tb, tg, tbt, ta, vW, vb, vg, vbt, va, X);

  // 2) Degrees + per-edge norm.
  zero_kernel<<<(N_NODES + 255) / 256, 256, 0, stream>>>(deg, N_NODES);
  deg_kernel<<<N_EDGES / 256, 256, 0, stream>>>(dst, deg);
  norm_kernel<<<N_EDGES / 256, 256, 0, stream>>>(src, dst, deg, nrm);

  // 3) K_DIFF=4 diffusion steps; final step folds in 1/z.
  const float bk4  = BETA_C * BETA_C * BETA_C * BETA_C;
  const float zf   = ALPHA_C * (1.0f - bk4) / (1.0f - BETA_C) + bk4;
  const float invz = 1.0f / zf;
  const int NH = N_NODES * HID;
  for (int it = 0; it < 4; ++it) {
    zero_kernel<<<NH / 256, 256, 0, stream>>>(tmp, NH);
    const float* hin = (it == 0) ? X : h;
    scatter_kernel<<<(N_EDGES / 4), 256, 0, stream>>>(hin, src, dst, nrm, tmp);
    axpy_kernel<<<NH / 256, 256, 0, stream>>>(X, tmp, h,
                                              (it == 3) ? invz : 1.0f);
  }

  // 4) Per-node MHA + residual + classifier (dynamic LDS ~70 KB).
  size_t mha_lds = (size_t)(4 * 16 * TSTRIDE + 4 * 16 * 16) * sizeof(float);
  mha_kernel<<<N_NODES, 256, mha_lds, stream>>>(
      h, memidx, Wq, bq, Wk, bk, Wv, bvp, Wo, bo, Wc, bc,
      out_z, out_final, out_logits);
}